// MuxoutTranspose_22969485099412
// MI455X (gfx1250) — compile-verified
//
#include <hip/hip_runtime.h>
#include <stdint.h>

// MuxoutTranspose: out[b, n*4+g_in, h, w] = sum_{g_out} x[b, g_out*16+n, 2h+sh, 2w+sw]
// with (sh,sw) = decode(iperm[g_out, g_in]).
// Streaming permute-reduce: 268 MB in + 67 MB out -> ~14.4 us floor at 23.3 TB/s.
// Data path: CDNA5 Tensor Data Mover (TENSORcnt) loads a 3D tile
// (4 channels x 2 rows x 512 f32, strided) straight into LDS with ONE
// instruction per block (issued by wave 0 only).

#define B_    4
#define C_    64
#define GOUT  4
#define GIN   4
#define NG    16
#define INH   512
#define INW   512
#define H_    256
#define W_    256

typedef float    v2f __attribute__((ext_vector_type(2)));
typedef uint32_t v4u __attribute__((ext_vector_type(4)));
typedef int      v8i __attribute__((ext_vector_type(8)));
typedef int      v4i __attribute__((ext_vector_type(4)));

__global__ __launch_bounds__(128) void muxout_tmux_kernel(
    const float* __restrict__ x,
    const int*   __restrict__ iperm,
    float*       __restrict__ out)
{
  __shared__ float tile[GOUT][2][INW];   // 16 KB: [g_out][sh][input col]  (= TDM [Z][Y][X])
  __shared__ int   s_ip[16];

  const int t   = threadIdx.x;           // 0..127
  const int bid = blockIdx.x;            // 0..16383
  const int b   = bid >> 12;             // / (NG*H_)
  const int rem = bid & 4095;
  const int n   = rem >> 8;              // / H_
  const int h   = rem & 255;

  if (t < 16) s_ip[t] = iperm[t];

  // Low 32 bits of the generic shared-pointer = workgroup-relative LDS byte offset.
  const uint32_t lds_base = (uint32_t)(uintptr_t)(&tile[0][0][0]);

#if __has_builtin(__builtin_amdgcn_tensor_load_to_lds)
  if (t < 32) {  // wave 0 only: a single DMA per block (branch skips other waves)
    const uint64_t ga = (uint64_t)(uintptr_t)
        (x + ((size_t)(b * C_ + n) * INH + (size_t)(2 * h)) * INW);  // chan g_out=0, row 2h, col 0

    // D# group 0: count=1 | lds_addr | global_addr[56:0] | type=2
    v4u g0 = { 1u,
               lds_base,
               (uint32_t)ga,
               ((uint32_t)(ga >> 32) & 0x01FFFFFFu) | (2u << 30) };

    // D# group 1:
    //  dw0: workgroup_mask=0, data_size=2 (4B), no barrier/iterate/pad
    //  tensor_dim0=512 @[79:48], tensor_dim1=2 @[111:80], tile_dim0=512 @[127:112]
    //  tile_dim1=2, tile_dim2=4 @[159:128]
    //  tensor_dim0_stride=512 (row stride) @[207:160]
    //  tensor_dim1_stride=NG*INH*INW=4194304 (channel stride) @[255:208] -> dw7=0x40
    v8i g1 = { (int)(2u << 16),
               (int)(512u << 16),            // tensor_dim0[15:0]
               (int)(2u << 16),              // tensor_dim0[31:16]=0 | tensor_dim1[15:0]=2
               (int)(512u << 16),            // tensor_dim1[31:16]=0 | tile_dim0=512
               (int)(2u | (4u << 16)),       // tile_dim1=2 | tile_dim2=4
               512,                          // tensor_dim0_stride[31:0]
               0,                            // stride0[47:32]=0 | stride1[15:0]=0
               0x40 };                       // tensor_dim1_stride[47:16] (0x400000>>16)

    // D# group 2: tensor_dim2=4, tensor_dim3=1, dim2_stride unused, tile_dim3=0
    v4i g2 = { 4, 1, 0, 0 };
    // D# group 3: dim3_stride=0, tensor_dim4=1 @[79:48], tile_dim4=0
    v4i g3 = { 0, (int)(1u << 16), 0, 0 };

#if defined(__clang_major__) && (__clang_major__ >= 23)
    // amdgpu-toolchain (clang-23): 6-arg form with extra (unused/reserved) v8i slot.
    v8i g4 = { 0, 0, 0, 0, 0, 0, 0, 0 };
    __builtin_amdgcn_tensor_load_to_lds(g0, g1, g2, g3, g4, 0);
#else
    // ROCm 7.2 (clang-22): 5-arg form.
    __builtin_amdgcn_tensor_load_to_lds(g0, g1, g2, g3, 0);
#endif
    __builtin_amdgcn_s_wait_tensorcnt(0);    // wave 0's DMA complete
  }
#else
  // Fallback: per-lane async global->LDS DMA (ASYNCcnt), 16B chunks, coalesced.
#pragma unroll
  for (int i = 0; i < 8; ++i) {
    const int c      = i * 128 + t;      // 0..1023 chunks of 16B
    const int row_id = c >> 7;           // g_out*2 + sh
    const int colc   = c & 127;
    const int go     = row_id >> 1;
    const int r      = row_id & 1;
    const size_t chan = (size_t)(b * C_ + go * NG + n);
    const float* g = x + (chan * INH + (size_t)(2 * h + r)) * INW + colc * 4;
    const uint32_t lofs = lds_base + (uint32_t)((((go * 2 + r) * INW) + colc * 4) * 4);
    asm volatile("global_load_async_to_lds_b128 %0, %1, off"
                 :: "v"(lofs), "v"((uint64_t)(uintptr_t)g)
                 : "memory");
  }
  asm volatile("s_wait_asynccnt 0x0" ::: "memory");
#endif

  __syncthreads();   // publish LDS tile (and s_ip) to all waves

  // Each thread: output pixels (h, 2t) and (h, 2t+1).
  const int wp = t;
  float4 q0[GOUT], q1[GOUT];
#pragma unroll
  for (int go = 0; go < GOUT; ++go) {
    q0[go] = *(const float4*)&tile[go][0][4 * wp];   // row 2h   : cols 4wp..4wp+3
    q1[go] = *(const float4*)&tile[go][1][4 * wp];   // row 2h+1
  }
  int ip[16];
#pragma unroll
  for (int i = 0; i < 16; ++i) ip[i] = s_ip[i];

#pragma unroll
  for (int gi = 0; gi < GIN; ++gi) {
    float a0 = 0.f, a1 = 0.f;
#pragma unroll
    for (int go = 0; go < GOUT; ++go) {
      const int p = ip[go * 4 + gi];                 // p = sh*2 + sw
      const float e0 = (p & 1) ? q0[go].y : q0[go].x;  // sh=0, pixel0
      const float e1 = (p & 1) ? q0[go].w : q0[go].z;  // sh=0, pixel1
      const float f0 = (p & 1) ? q1[go].y : q1[go].x;  // sh=1, pixel0
      const float f1 = (p & 1) ? q1[go].w : q1[go].z;  // sh=1, pixel1
      a0 += (p & 2) ? f0 : e0;
      a1 += (p & 2) ? f1 : e1;
    }
    float* dst = out + (((size_t)(b * C_ + n * GIN + gi) * H_) + h) * W_ + 2 * wp;
    v2f st = {a0, a1};
    __builtin_nontemporal_store(st, (v2f*)dst);      // streaming store, no reuse
  }
}

extern "C" void kernel_launch(void* const* d_in, const int* in_sizes, int n_in,
                              void* d_out, int out_size, void* d_ws, size_t ws_size,
                              hipStream_t stream) {
  (void)in_sizes; (void)n_in; (void)d_ws; (void)ws_size; (void)out_size;
  const float* x     = (const float*)d_in[0];
  const int*   iperm = (const int*)d_in[1];
  float*       out   = (float*)d_out;
  const int blocks = B_ * NG * H_;   // 4 * 16 * 256 = 16384
  muxout_tmux_kernel<<<blocks, 128, 0, stream>>>(x, iperm, out);
}